// RedistributionRefiner_28638841930125
// MI455X (gfx1250) — compile-verified
//
#include <hip/hip_runtime.h>
#include <stdint.h>

// RedistributionRefiner reduces to: out[b, j, :] = r[b, idx_unbiased[b, j], :]
// (merged[:, :k] == ru exactly; all matmul/relu work is dead code).
// Pure row gather: 8*2048 rows of 256 f32 (1 KB each) -> ~32 MB traffic,
// memory-bound (~1.4 us at 23.3 TB/s). Implemented with the CDNA5 Tensor
// Data Mover in GATHER mode: each descriptor gathers 8 rows by 32-bit index
// into LDS; a normal-mode 2D descriptor streams the contiguous result out.

namespace {
constexpr int Bb = 8;
constexpr int Nn = 4096;
constexpr int Dd = 256;
constexpr int Kk = 2048;
constexpr int ROWS_PER_DESC = 8;                 // 32-bit gather indices: max 8 rows / D#
constexpr int DESCS = 2;
constexpr int ROWS = ROWS_PER_DESC * DESCS;      // 16 rows per workgroup
constexpr int ROW_BYTES = Dd * 4;                // 1 KB per row
}

typedef unsigned int u32x4 __attribute__((ext_vector_type(4)));
typedef int          i32x4 __attribute__((ext_vector_type(4)));
typedef int          i32x8 __attribute__((ext_vector_type(8)));

#if __has_builtin(__builtin_amdgcn_tensor_load_to_lds) && \
    __has_builtin(__builtin_amdgcn_tensor_store_from_lds)
#define HAVE_TDM 1
#else
#define HAVE_TDM 0
#endif

#if HAVE_TDM
__device__ __forceinline__ void tdm_load(u32x4 g0, i32x8 g1, i32x4 g2, i32x4 g3) {
#if defined(__clang_major__) && (__clang_major__ >= 23)
  i32x8 gz = {};
  __builtin_amdgcn_tensor_load_to_lds(g0, g1, g2, g3, gz, 0);
#else
  __builtin_amdgcn_tensor_load_to_lds(g0, g1, g2, g3, 0);
#endif
}
__device__ __forceinline__ void tdm_store(u32x4 g0, i32x8 g1, i32x4 g2, i32x4 g3) {
#if defined(__clang_major__) && (__clang_major__ >= 23)
  i32x8 gz = {};
  __builtin_amdgcn_tensor_store_from_lds(g0, g1, g2, g3, gz, 0);
#else
  __builtin_amdgcn_tensor_store_from_lds(g0, g1, g2, g3, 0);
#endif
}
#endif

__global__ __launch_bounds__(32)
void RedistributionRefiner_gather_tdm(const float* __restrict__ r,
                                      const int* __restrict__ idx,
                                      float* __restrict__ out) {
  __shared__ float smem[ROWS * Dd];   // 16 KB staging

  const uint32_t blocks_per_batch = Kk / ROWS;                 // 128
  const uint32_t b  = (uint32_t)blockIdx.x / blocks_per_batch;
  const uint32_t j0 = ((uint32_t)blockIdx.x % blocks_per_batch) * ROWS;

  const int* ip = idx + (size_t)b * Kk + j0;                   // uniform -> SGPRs

#if HAVE_TDM
  const uint32_t lds_base = (uint32_t)(uintptr_t)(&smem[0]);   // low 32 bits = LDS offset
  const uint64_t src = (uint64_t)(uintptr_t)(r + (uint64_t)b * Nn * Dd);

#pragma unroll
  for (int g = 0; g < DESCS; ++g) {
    u32x4 g0; i32x8 g1; i32x4 g2, g3;
    // D# group 0: count=1 | 32-bit indices | gather mode; LDS dst; global src; type=2
    g0[0] = 1u | (1u << 30) | (1u << 31);
    g0[1] = lds_base + (uint32_t)g * ROWS_PER_DESC * ROW_BYTES;
    g0[2] = (uint32_t)src;
    g0[3] = ((uint32_t)(src >> 32) & 0x01FFFFFFu) | (2u << 30);

    // D# group 1
    g1[0] = (int)(2u << 16);                 // data_size = 4 bytes; no multicast/pad
    g1[1] = (int)((uint32_t)Dd << 16);       // tensor_dim0 = 256 elements (bits 63:48)
    g1[2] = (int)((uint32_t)Nn << 16);       // tensor_dim1 = 4096 rows   (bits 95:80)
    g1[3] = (int)((uint32_t)Dd << 16);       // tile_dim0   = 256         (bits 127:112)
    g1[4] = ROWS_PER_DESC;                   // tile_dim1   = #valid gather indices
    g1[5] = Dd;                              // tensor_dim0_stride = 256 elements
    g1[6] = 0;
    g1[7] = 0;

    // D# groups 2/3: 8 x 32-bit row indices
    const int* q = ip + g * ROWS_PER_DESC;
    g2[0] = q[0]; g2[1] = q[1]; g2[2] = q[2]; g2[3] = q[3];
    g3[0] = q[4]; g3[1] = q[5]; g3[2] = q[6]; g3[3] = q[7];

    tdm_load(g0, g1, g2, g3);
  }

  __builtin_amdgcn_s_wait_tensorcnt(0);      // gathers landed in LDS

  {
    // Contiguous 256x16 f32 tile LDS -> out (normal mode, 2D, tile == tensor)
    const uint64_t dst = (uint64_t)(uintptr_t)(out + ((uint64_t)b * Kk + j0) * Dd);
    u32x4 g0; i32x8 g1; i32x4 gz = {};
    g0[0] = 1u;                              // count=1, normal mode
    g0[1] = lds_base;
    g0[2] = (uint32_t)dst;
    g0[3] = ((uint32_t)(dst >> 32) & 0x01FFFFFFu) | (2u << 30);

    g1[0] = (int)(2u << 16);                 // data_size = 4 bytes
    g1[1] = (int)((uint32_t)Dd << 16);       // tensor_dim0 = 256
    g1[2] = (int)((uint32_t)ROWS << 16);     // tensor_dim1 = 16 rows
    g1[3] = (int)((uint32_t)Dd << 16);       // tile_dim0 = 256
    g1[4] = ROWS;                            // tile_dim1 = 16
    g1[5] = Dd;                              // tensor_dim0_stride = 256
    g1[6] = 0;
    g1[7] = 0;

    tdm_store(g0, g1, gz, gz);
  }
  // S_ENDPGM performs an implicit wait-idle -> store completion guaranteed.
#else
  // Fallback (toolchain without TDM builtins): direct vectorized row copy.
  for (int row = 0; row < ROWS; ++row) {
    const int srow = ip[row];
    const float4* s = (const float4*)(r + (uint64_t)b * Nn * Dd + (uint64_t)srow * Dd);
    float4* o = (float4*)(out + ((uint64_t)b * Kk + j0 + row) * Dd);
    for (int t = threadIdx.x; t < Dd / 4; t += 32) o[t] = s[t];
  }
#endif
}

extern "C" void kernel_launch(void* const* d_in, const int* in_sizes, int n_in,
                              void* d_out, int out_size, void* d_ws, size_t ws_size,
                              hipStream_t stream) {
  (void)in_sizes; (void)n_in; (void)out_size; (void)d_ws; (void)ws_size;
  const float* r            = (const float*)d_in[0];  // (B, N, D) f32
  const int*   idx_unbiased = (const int*)d_in[2];    // (B, K) i32
  float*       out          = (float*)d_out;          // (B, K, D) f32

  const int blocks = Bb * Kk / ROWS;                  // 1024 workgroups, 1 wave each
  RedistributionRefiner_gather_tdm<<<dim3(blocks), dim3(32), 0, stream>>>(
      r, idx_unbiased, out);
}